// BranchingAngleModule_85856396247190
// MI455X (gfx1250) — compile-verified
//
#include <hip/hip_runtime.h>
#include <hip/hip_bf16.h>
#include <stdint.h>

#define FEAT 512
#define SENTINEL 0x7fffffff

// ---------------------------------------------------------------------------
// Kernel 1: initialize segment-min arrays to INT_MAX (matches jax segment_min
// empty-segment sentinel).
// ---------------------------------------------------------------------------
__global__ void init_mins_kernel(int* __restrict__ n1, int* __restrict__ n2, int n) {
    int i = blockIdx.x * blockDim.x + threadIdx.x;
    if (i < n) {
        n1[i] = SENTINEL;
        n2[i] = SENTINEL;
    }
}

// ---------------------------------------------------------------------------
// Kernel 2: first segment-min (smallest-index neighbor), undirected edges.
// Non-returning global_atomic_min_i32 — scatter-reduce resolved at L2.
// ---------------------------------------------------------------------------
__global__ void min1_kernel(const int* __restrict__ src, const int* __restrict__ dst,
                            int* __restrict__ n1, int nE) {
    int i = blockIdx.x * blockDim.x + threadIdx.x;
    if (i >= nE) return;
    int s = src[i];
    int d = dst[i];
    atomicMin(&n1[s], d);
    atomicMin(&n1[d], s);
}

// ---------------------------------------------------------------------------
// Kernel 3: second segment-min, masking every copy of n1 (dedup semantics of
// the reference: nbr == n1[seg] -> num_nodes sentinel).
// ---------------------------------------------------------------------------
__global__ void min2_kernel(const int* __restrict__ src, const int* __restrict__ dst,
                            const int* __restrict__ n1, int* __restrict__ n2,
                            int nE, int N) {
    int i = blockIdx.x * blockDim.x + threadIdx.x;
    if (i >= nE) return;
    int s = src[i];
    int d = dst[i];
    int vd = (d == n1[s]) ? N : d;
    int vs = (s == n1[d]) ? N : s;
    atomicMin(&n2[s], vd);
    atomicMin(&n2[d], vs);
}

// ---------------------------------------------------------------------------
// Kernel 4: per-bifurcation angle score / violation (10K lanes, trivial).
// ---------------------------------------------------------------------------
__global__ void bifurc_kernel(const int* __restrict__ bidx,
                              const int* __restrict__ n1, const int* __restrict__ n2,
                              const float* __restrict__ pos,
                              float* __restrict__ scores, float* __restrict__ viols,
                              int B, int N) {
    int i = blockIdx.x * blockDim.x + threadIdx.x;
    if (i >= B) return;
    int b  = bidx[i];
    int c1 = n1[b];
    int c2 = n2[b];
    bool valid = (c1 < N) && (c2 < N);
    float score = 1.0f;
    float viol  = 0.0f;
    if (valid) {
        float px = pos[3 * b + 0], py = pos[3 * b + 1], pz = pos[3 * b + 2];
        float ax = pos[3 * c1 + 0] - px;
        float ay = pos[3 * c1 + 1] - py;
        float az = pos[3 * c1 + 2] - pz;
        float bx = pos[3 * c2 + 0] - px;
        float by = pos[3 * c2 + 1] - py;
        float bz = pos[3 * c2 + 2] - pz;
        float dot = ax * bx + ay * by + az * bz;
        float na  = sqrtf(ax * ax + ay * ay + az * az);
        float nb  = sqrtf(bx * bx + by * by + bz * bz);
        float cosv = dot / (na * nb);
        cosv = fminf(1.0f, fmaxf(-1.0f, cosv));
        float ang = acosf(cosv) * 57.29577951308232f;   // degrees
        bool  inside = (ang >= 30.0f) && (ang <= 60.0f);
        float dist = (ang < 30.0f) ? (30.0f - ang) : (ang - 60.0f);
        float score_out = fmaxf(0.0f, 1.0f - dist * (1.0f / 30.0f));
        score = inside ? 1.0f : score_out;
        viol  = inside ? 0.0f : (1.0f - score_out);
    }
    scores[i] = score;
    viols[i]  = viol;
}

// ---------------------------------------------------------------------------
// Kernel 5: dominant cost — 410 MB streaming copy of node_features via the
// CDNA5 async global<->LDS DMA path (ASYNCcnt).
//
// Pipelining: two LDS banks of 4 slots per thread. Per batch, issue 4 async
// loads (B128, fully coalesced: elements strided by gridDim*blockDim) into
// bank b while the 4 async stores from bank 1-b are still outstanding, then
// s_wait_asynccnt 0 (the only safe threshold: ASYNCcnt mixes loads+stores and
// they complete cross-type out of order), then issue 4 async stores.
// NT temporal hints both ways: the 205 MB/direction stream exceeds the 192 MB
// L2 and has zero reuse, so keep it from sweeping the cache.
// LDS: 2 banks * 4 slots * 256 thr * 16 B = 32 KB/block (<< 320 KB/WGP).
// ---------------------------------------------------------------------------
__global__ void copy_features_async_kernel(const float4* __restrict__ src,
                                           float4* __restrict__ dst, long n4) {
    __shared__ float4 smem[2][4][256];
    const int tid = threadIdx.x;
    uint32_t lds[2][4];
#pragma unroll
    for (int b = 0; b < 2; ++b)
#pragma unroll
        for (int k = 0; k < 4; ++k)
            // addrspace(3) flat address: low 32 bits are the LDS byte offset
            lds[b][k] = (uint32_t)(uintptr_t)(&smem[b][k][tid]);

    const long S = (long)gridDim.x * blockDim.x;      // coalescing stride
    long i = (long)blockIdx.x * blockDim.x + tid;
    int bank = 0;

    // Batched main loop: 4 outstanding async loads per wave between waits.
    for (; i + 3 * S < n4; i += 4 * S) {
#pragma unroll
        for (int k = 0; k < 4; ++k) {
            unsigned long long ga = (unsigned long long)(uintptr_t)(src + i + k * S);
            asm volatile("global_load_async_to_lds_b128 %0, %1, off th:TH_LOAD_NT"
                         :: "v"(lds[0][k] + (bank ? (uint32_t)(4 * 256 * 16) : 0u)),
                            "v"(ga) : "memory");
        }
        // Drain these 4 loads (and the previous bank's 4 stores).
        asm volatile("s_wait_asynccnt 0" ::: "memory");
#pragma unroll
        for (int k = 0; k < 4; ++k) {
            unsigned long long gd = (unsigned long long)(uintptr_t)(dst + i + k * S);
            asm volatile("global_store_async_from_lds_b128 %0, %1, off th:TH_STORE_NT"
                         :: "v"(gd),
                            "v"(lds[0][k] + (bank ? (uint32_t)(4 * 256 * 16) : 0u))
                         : "memory");
        }
        bank ^= 1;
    }
    // Tail: one element at a time with bounds check.
    for (; i < n4; i += S) {
        unsigned long long ga = (unsigned long long)(uintptr_t)(src + i);
        unsigned long long gd = (unsigned long long)(uintptr_t)(dst + i);
        asm volatile("global_load_async_to_lds_b128 %0, %1, off th:TH_LOAD_NT"
                     :: "v"(lds[0][0]), "v"(ga) : "memory");
        asm volatile("s_wait_asynccnt 0" ::: "memory");
        asm volatile("global_store_async_from_lds_b128 %0, %1, off th:TH_STORE_NT"
                     :: "v"(gd), "v"(lds[0][0]) : "memory");
    }
    // Drain all remaining async stores before wave exit.
    asm volatile("s_wait_asynccnt 0" ::: "memory");
}

// ---------------------------------------------------------------------------
// Launch. Inputs (setup_inputs order):
//   d_in[0] node_features  f32 [100000*512]
//   d_in[1] edge_index     i32 [2*3200000]   (row0 = src, row1 = dst)
//   d_in[2] node_positions f32 [100000*3]
//   d_in[3] bifurc_idx     i32 [10000]
// d_out = [features copy | angle_scores | angle_violations]
// ---------------------------------------------------------------------------
extern "C" void kernel_launch(void* const* d_in, const int* in_sizes, int n_in,
                              void* d_out, int out_size, void* d_ws, size_t ws_size,
                              hipStream_t stream) {
    const float* feats = (const float*)d_in[0];
    const int*   eidx  = (const int*)d_in[1];
    const float* pos   = (const float*)d_in[2];
    const int*   bidx  = (const int*)d_in[3];

    const int nNodes = in_sizes[0] / FEAT;     // 100000
    const int nE     = in_sizes[1] / 2;        // 3200000
    const int B      = in_sizes[3];            // 10000

    const int* src = eidx;
    const int* dst = eidx + nE;

    int* n1 = (int*)d_ws;                      // nNodes ints
    int* n2 = n1 + nNodes;                     // nNodes ints   (800 KB total)

    float* out    = (float*)d_out;
    float* scores = out + (size_t)nNodes * FEAT;
    float* viols  = scores + B;

    init_mins_kernel<<<(nNodes + 255) / 256, 256, 0, stream>>>(n1, n2, nNodes);
    min1_kernel<<<(nE + 255) / 256, 256, 0, stream>>>(src, dst, n1, nE);
    min2_kernel<<<(nE + 255) / 256, 256, 0, stream>>>(src, dst, n1, n2, nE, nNodes);
    bifurc_kernel<<<(B + 255) / 256, 256, 0, stream>>>(bidx, n1, n2, pos, scores, viols, B, nNodes);

    const long n4 = (long)nNodes * FEAT / 4;   // 12.8M float4s
    copy_features_async_kernel<<<4096, 256, 0, stream>>>(
        (const float4*)feats, (float4*)out, n4);
}